// StickBreaking_50345606643969
// MI455X (gfx1250) — compile-verified
//
#include <hip/hip_runtime.h>
#include <cstdint>

// ---------------------------------------------------------------------------
// Stick-breaking over sigmoid(x), [B=256, N=256, N=256], fp32.
// Latency-bound serial scan (N*N steps/batch, only B-way parallel).
// CDNA5 path: ALL global traffic via Tensor Data Mover.
//   - loads : strided [32 x 256] row tiles -> padded LDS (stride 257, via
//             TDM pad_interval=256dw/pad_amount=1dw) => bank-conflict-free.
//   - stores: unpadded [32 x 256] LDS tiles -> strided global rows.
// Double-buffered both ways on TENSORcnt; no barriers (1 wave / block,
// zero cross-lane LDS sharing). Critical chain is depth-4 VALU:
//   M = max(a_n, r); m' = min(1, c_n + r); r' = fma(b, m'-M, M); p = r'-r.
// Suffix sums maintained as S[n] = Ttot - PS[n], with PS[n] += r' updated
// inline in the forward scan -> no backward pass at all.
// ---------------------------------------------------------------------------

typedef unsigned int u32x4 __attribute__((ext_vector_type(4)));
typedef int          i32x8 __attribute__((ext_vector_type(8)));
typedef int          i32x4 __attribute__((ext_vector_type(4)));

#define NN     256   // N (matrix dim)
#define NB     32    // batches per block == wave32
#define STRIDE 257   // padded LDS row stride (floats)

__device__ __forceinline__ void wait_dscnt0() {
#if __has_builtin(__builtin_amdgcn_s_wait_dscnt)
  __builtin_amdgcn_s_wait_dscnt(0);
#else
  asm volatile("s_wait_dscnt 0" ::: "memory");
#endif
}

// TDM load: 2D tile [NB x NN] f32, batch stride NN*NN, into padded LDS.
__device__ __forceinline__ void tdm_load_row_tile(const float* gsrc, uint32_t lds_off) {
  uint64_t ga = (uint64_t)(uintptr_t)gsrc;
  u32x4 g0;
  g0[0] = 1u;                                                      // count=1
  g0[1] = lds_off;                                                 // lds_addr
  g0[2] = (uint32_t)ga;                                            // global_addr lo
  g0[3] = (uint32_t)((ga >> 32) & 0x01FFFFFFu) | 0x80000000u;      // ga hi | type=2
  i32x8 g1;
  g1[0] = 0x01D20000;  // data_size=4B, pad_enable, pad_interval=7(256dw), pad_amount=0(1dw)
  g1[1] = 0x01000000;  // tensor_dim0 = 256
  g1[2] = 0x00200000;  // tensor_dim1 = 32
  g1[3] = 0x01000000;  // tile_dim0   = 256
  g1[4] = 32;          // tile_dim1   = 32, tile_dim2 = 0
  g1[5] = 0x00010000;  // tensor_dim0_stride = 65536 elements
  g1[6] = 0;  g1[7] = 0;
  i32x4 z4 = {0, 0, 0, 0};
#if __clang_major__ >= 23
  i32x8 z8 = {0, 0, 0, 0, 0, 0, 0, 0};
  __builtin_amdgcn_tensor_load_to_lds(g0, g1, z4, z4, z8, 0);
#else
  __builtin_amdgcn_tensor_load_to_lds(g0, g1, z4, z4, 0);
#endif
}

// TDM store: unpadded [NB x NN] f32 LDS tile -> global, batch stride NN*NN.
__device__ __forceinline__ void tdm_store_row_tile(float* gdst, uint32_t lds_off) {
  uint64_t ga = (uint64_t)(uintptr_t)gdst;
  u32x4 g0;
  g0[0] = 1u;
  g0[1] = lds_off;
  g0[2] = (uint32_t)ga;
  g0[3] = (uint32_t)((ga >> 32) & 0x01FFFFFFu) | 0x80000000u;
  i32x8 g1;
  g1[0] = 0x00020000;  // data_size=4B, no padding
  g1[1] = 0x01000000;  // tensor_dim0 = 256
  g1[2] = 0x00200000;  // tensor_dim1 = 32
  g1[3] = 0x01000000;  // tile_dim0   = 256
  g1[4] = 32;          // tile_dim1   = 32
  g1[5] = 0x00010000;  // tensor_dim0_stride = 65536
  g1[6] = 0;  g1[7] = 0;
  i32x4 z4 = {0, 0, 0, 0};
#if __clang_major__ >= 23
  i32x8 z8 = {0, 0, 0, 0, 0, 0, 0, 0};
  __builtin_amdgcn_tensor_store_from_lds(g0, g1, z4, z4, z8, 0);
#else
  __builtin_amdgcn_tensor_store_from_lds(g0, g1, z4, z4, 0);
#endif
}

__global__ __launch_bounds__(NB) void stick_breaking_kernel(
    const float* __restrict__ in, float* __restrict__ out) {
  extern __shared__ float smem[];
  float* tile0 = smem;                        // 32*257 padded input buffer A
  float* tile1 = smem + NB * STRIDE;          // 32*257 padded input buffer B
  float* st0   = smem + 2 * NB * STRIDE;      // 32*256 unpadded store buffer A
  float* st1   = st0 + NB * NN;               // 32*256 unpadded store buffer B
  float* Carr  = st1 + NB * NN;               // 32*257 column sums C[n]
  float* PSarr = Carr + NB * STRIDE;          // 32*257 summed prefix sums PS[n]

  const int t          = threadIdx.x;         // lane = batch within block
  const int batch_base = blockIdx.x * NB;

  const float* gin  = in  + (size_t)batch_base * (NN * NN);
  float*       gout = out + (size_t)batch_base * (NN * NN);
  const uint32_t lds_t0 = (uint32_t)(uintptr_t)tile0;
  const uint32_t lds_t1 = (uint32_t)(uintptr_t)tile1;
  const uint32_t lds_s0 = (uint32_t)(uintptr_t)st0;
  const uint32_t lds_s1 = (uint32_t)(uintptr_t)st1;

  // Prime: rows 0 and 1 in flight immediately (overlaps with C/PS init).
  tdm_load_row_tile(gin, lds_t0);
  tdm_load_row_tile(gin + (size_t)NN, lds_t1);

  // Lane-private init (no cross-lane LDS sharing anywhere in this kernel).
  float* Crow  = Carr  + t * STRIDE;
  float* PSrow = PSarr + t * STRIDE;
  #pragma unroll 8
  for (int n = 0; n < NN; ++n) { Crow[n] = 0.0f; PSrow[n] = 0.0f; }

  float Ttot = 0.0f;  // total mass of all previous rows (register)

  for (int m = 0; m < NN; ++m) {
    // In-order TENSORcnt: "<=2" leaves only {S(m-1), L(m+1)} outstanding,
    // so L(m) is complete and S(m-2) (this iteration's store buffer) drained.
    if (m == 0 || m == NN - 1) __builtin_amdgcn_s_wait_tensorcnt(1);
    else                       __builtin_amdgcn_s_wait_tensorcnt(2);
    asm volatile("" ::: "memory");

    const float* xrow = ((m & 1) ? tile1 : tile0) + t * STRIDE;  // padded, read-only
    float*       srow = ((m & 1) ? st1   : st0)   + t * NN;      // TDM store source

    // ---- forward serial scan (true dependency chain: depth-4 VALU) ----
    float r = 0.0f;
    #pragma unroll 8
    for (int n = 0; n < NN; ++n) {
      float x   = xrow[n];
      float b   = __fdividef(1.0f, 1.0f + __expf(-x));   // sigmoid       (off-chain)
      float PSn = PSrow[n];
      float an  = (float)(n - (NN - 2)) + (Ttot - PSn);  // 2-N+n+suffix  (off-chain)
      float cn  = 1.0f - Crow[n];                        //               (off-chain)
      float M   = fmaxf(an, r);                          // lb = M - r
      float mp  = fminf(1.0f, cn + r);                   // ub = mp - r
      float rn  = fmaf(b, mp - M, M);                    // r' = r + p
      float p   = rn - r;
      srow[n]  = p;                                      // output staging (off-chain)
      Crow[n] += p;                                      // column sums    (off-chain)
      PSrow[n] = PSn + rn;                               // += prefix sum  (off-chain)
      r = rn;
    }
    Ttot += r;  // S[n] for the next row = Ttot - PS[n]

    // ---- DMA this row out, prefetch row m+2 into the freed padded buffer ----
    asm volatile("" ::: "memory");
    wait_dscnt0();  // p-stores visible in LDS before TDM reads them
    tdm_store_row_tile(gout + (size_t)m * NN, (m & 1) ? lds_s1 : lds_s0);
    if (m + 2 < NN)
      tdm_load_row_tile(gin + (size_t)(m + 2) * NN, (m & 1) ? lds_t1 : lds_t0);
  }

  __builtin_amdgcn_s_wait_tensorcnt(0);  // flush trailing stores
}

extern "C" void kernel_launch(void* const* d_in, const int* in_sizes, int n_in,
                              void* d_out, int out_size, void* d_ws, size_t ws_size,
                              hipStream_t stream) {
  const float* in  = (const float*)d_in[0];
  float*       out = (float*)d_out;

  const int total  = in_sizes[0];          // B*N*N
  const int B      = total / (NN * NN);    // 256
  const int blocks = (B + NB - 1) / NB;    // 8

  // 2*(32*257) + 2*(32*256) + 2*(32*257) floats = 197,120 B (< 320 KB/WGP)
  const size_t smem_bytes =
      ((size_t)4 * NB * STRIDE + (size_t)2 * NB * NN) * sizeof(float);
  hipFuncSetAttribute((const void*)stick_breaking_kernel,
                      hipFuncAttributeMaxDynamicSharedMemorySize, (int)smem_bytes);

  stick_breaking_kernel<<<blocks, NB, smem_bytes, stream>>>(in, out);
}